// ROITimeTransformer_76605036692268
// MI455X (gfx1250) — compile-verified
//
#include <hip/hip_runtime.h>
#include <hip/hip_bf16.h>
#include <math.h>

// Problem dims (fixed by the reference)
#define B_      4
#define C_      256
#define T_      8
#define H_      64
#define W_      64
#define N_      16
#define P_      4
#define DM_     128
#define NHEAD_  8
#define DH_     16
#define RSCALE  (1.0f/16.0f)

typedef __attribute__((ext_vector_type(2))) float v2f;
typedef __attribute__((ext_vector_type(8))) float v8f;

#if defined(__gfx1250__) && __has_builtin(__builtin_amdgcn_wmma_f32_16x16x4_f32)
#define HAVE_WMMA_F32X4 1
#else
#define HAVE_WMMA_F32X4 0
#endif

// ---------------------------------------------------------------------------
// 1) ROI max-pool: x [B,C,T,H,W] + boxes [B,T,N,4] -> pooled [8192 x 256]
//    row = ((b*T+t)*N + n)*16 + (ph*4+pw), col = channel  (GEMM-ready layout)
// ---------------------------------------------------------------------------
__global__ __launch_bounds__(256)
void roipool_kernel(const float* __restrict__ x, const float* __restrict__ boxes,
                    float* __restrict__ pooled)
{
    int r  = blockIdx.x;          // 0..511  = (b*T+t)*N + n
    int c  = threadIdx.x;         // channel
    int bt = r / N_;
    int b  = bt / T_, t = bt % T_;

    float x1 = boxes[r*4+0], y1 = boxes[r*4+1];
    float x2 = boxes[r*4+2], y2 = boxes[r*4+3];
    float sw = roundf(x1 * RSCALE), sh = roundf(y1 * RSCALE);
    float ew = roundf(x2 * RSCALE), eh = roundf(y2 * RSCALE);
    float rw = fmaxf(ew - sw + 1.0f, 1.0f);
    float rh = fmaxf(eh - sh + 1.0f, 1.0f);

    int wsA[P_], weA[P_], hsA[P_], heA[P_];
    for (int p = 0; p < P_; ++p) {
        wsA[p] = (int)fminf(fmaxf(floorf((float)p      * rw * (1.0f/P_)) + sw, 0.f), (float)W_);
        weA[p] = (int)fminf(fmaxf(ceilf ((float)(p+1)  * rw * (1.0f/P_)) + sw, 0.f), (float)W_);
        hsA[p] = (int)fminf(fmaxf(floorf((float)p      * rh * (1.0f/P_)) + sh, 0.f), (float)H_);
        heA[p] = (int)fminf(fmaxf(ceilf ((float)(p+1)  * rh * (1.0f/P_)) + sh, 0.f), (float)H_);
    }

    const float* f = x + ((size_t)(b*C_ + c)*T_ + t) * (size_t)(H_*W_);
    for (int ph = 0; ph < P_; ++ph) {
        for (int pw = 0; pw < P_; ++pw) {
            bool any = (heA[ph] > hsA[ph]) && (weA[pw] > wsA[pw]);
            float m = -1e30f;
            for (int h = hsA[ph]; h < heA[ph]; ++h)
                for (int w = wsA[pw]; w < weA[pw]; ++w)
                    m = fmaxf(m, f[h*W_ + w]);
            pooled[((size_t)r*16 + ph*4 + pw)*C_ + c] = any ? m : 0.0f;
        }
    }
}

// ---------------------------------------------------------------------------
// 2) Generic f32 WMMA GEMM:  C[M,N] = act(A[M,K] * W[N,K]^T + bias)
//    One wave (32 lanes) per 16x16 output tile, K-loop in steps of 4 using
//    V_WMMA_F32_16X16X4_F32.  ISA lane layouts:
//      A 16x4 : lane L (m=L&15, half=L>>4), VGPR j -> A[m][k0 + 2*half + j]
//      B 4x16 : lane L (n=L&15, half=L>>4), VGPR j -> B[k0 + 2*half + j][n]
//      C 16x16: lane L, VGPR v -> row (v + 8*half), col (L&15)
//    WMODE==1: contraction index p of A corresponds to past_w column
//              (p&127)*16 + (p>>7)  (the k*16+bin flatten permutation).
//    ACT: 0=none, 1=relu, 2=elu
// ---------------------------------------------------------------------------
template<int ACT, int WMODE>
__global__ __launch_bounds__(32)
void gemm16(const float* __restrict__ A, int lda,
            const float* __restrict__ Wt, int ldw,
            const float* __restrict__ bias,
            float* __restrict__ Co, int ldc,
            int M, int Nn, int K)
{
    const int tilesM = M >> 4;
    const int tM   = (blockIdx.x % tilesM) << 4;
    const int tN   = (blockIdx.x / tilesM) << 4;
    const int lane = threadIdx.x;
    const int half = lane >> 4;
    const int lm   = lane & 15;

    const float* arow = A  + (size_t)(tM + lm) * lda;
    const float* wrow = Wt + (size_t)(tN + lm) * (WMODE == 0 ? ldw : 2048);

#if HAVE_WMMA_F32X4
    v8f acc = {};
    for (int k0 = 0; k0 < K; k0 += 4) {
        const int ka = k0 + half*2;
        v2f a; a.x = arow[ka]; a.y = arow[ka + 1];
        v2f bm;
        if (WMODE == 0) {
            bm.x = wrow[ka]; bm.y = wrow[ka + 1];
        } else {
            const int p0 = ka, p1 = ka + 1;
            bm.x = wrow[(p0 & 127)*16 + (p0 >> 7)];
            bm.y = wrow[(p1 & 127)*16 + (p1 >> 7)];
        }
        acc = __builtin_amdgcn_wmma_f32_16x16x4_f32(false, a, false, bm,
                                                    (short)0, acc, false, false);
    }
    const float bv = bias ? bias[tN + lm] : 0.0f;
    for (int v = 0; v < 8; ++v) {
        const int row = tM + v + half*8;
        float val = acc[v] + bv;
        if (ACT == 1) val = fmaxf(val, 0.0f);
        if (ACT == 2) val = val > 0.0f ? val : (expf(val) - 1.0f);
        Co[(size_t)row*ldc + tN + lm] = val;
    }
#else
    // Scalar fallback (host pass / missing builtin): same tile mapping.
    float acc[8] = {0,0,0,0,0,0,0,0};
    for (int k = 0; k < K; ++k) {
        float wv = (WMODE == 0) ? wrow[k] : wrow[(k & 127)*16 + (k >> 7)];
        for (int v = 0; v < 8; ++v)
            acc[v] += A[(size_t)(tM + v + half*8)*lda + k] * wv;
    }
    const float bv = bias ? bias[tN + lm] : 0.0f;
    for (int v = 0; v < 8; ++v) {
        const int row = tM + v + half*8;
        float val = acc[v] + bv;
        if (ACT == 1) val = fmaxf(val, 0.0f);
        if (ACT == 2) val = val > 0.0f ? val : (expf(val) - 1.0f);
        Co[(size_t)row*ldc + tN + lm] = val;
    }
#endif
}

// ---------------------------------------------------------------------------
// 3) Attention: one block per (batch, head); K/V staged in LDS; one thread
//    per query token (Lq=256) with single-pass online softmax, dh=16.
// ---------------------------------------------------------------------------
__global__ __launch_bounds__(256)
void attn_kernel(const float* __restrict__ Q, int ldq,
                 const float* __restrict__ Kp, const float* __restrict__ Vp, int ldkv,
                 float* __restrict__ O, int ldo, int Lk)
{
    __shared__ float sK[256*DH_];
    __shared__ float sV[256*DH_];
    const int b   = blockIdx.x / NHEAD_;
    const int h   = blockIdx.x % NHEAD_;
    const int tid = threadIdx.x;

    for (int idx = tid; idx < Lk*DH_; idx += 256) {
        const int j = idx >> 4, d = idx & 15;
        sK[idx] = Kp[(size_t)(b*Lk + j)*ldkv + h*DH_ + d];
        sV[idx] = Vp[(size_t)(b*Lk + j)*ldkv + h*DH_ + d];
    }
    __syncthreads();

    const int i = tid;  // query 0..255
    float qr[DH_];
    const float* qp = Q + (size_t)(b*256 + i)*ldq + h*DH_;
    for (int d = 0; d < DH_; ++d) qr[d] = qp[d];

    float mx = -1e30f, sum = 0.0f, acc[DH_];
    for (int d = 0; d < DH_; ++d) acc[d] = 0.0f;
    const float sc = 0.25f;  // 1/sqrt(16)

    for (int j = 0; j < Lk; ++j) {
        float s = 0.0f;
        for (int d = 0; d < DH_; ++d) s += qr[d] * sK[j*DH_ + d];
        s *= sc;
        if (s > mx) {
            const float f = expf(mx - s);
            sum *= f;
            for (int d = 0; d < DH_; ++d) acc[d] *= f;
            mx = s;
        }
        const float w = expf(s - mx);
        sum += w;
        for (int d = 0; d < DH_; ++d) acc[d] += w * sV[j*DH_ + d];
    }
    const float inv = 1.0f / sum;
    float* op = O + (size_t)(b*256 + i)*ldo + h*DH_;
    for (int d = 0; d < DH_; ++d) op[d] = acc[d] * inv;
}

// ---------------------------------------------------------------------------
// 4) Fused residual + LayerNorm over D=128; one 128-thread block per row.
// ---------------------------------------------------------------------------
__global__ __launch_bounds__(128)
void add_ln_kernel(const float* __restrict__ a, const float* __restrict__ bsrc,
                   const float* __restrict__ g, const float* __restrict__ beta,
                   float* __restrict__ out)
{
    __shared__ float rs[128];
    __shared__ float stat;
    const int row = blockIdx.x, tid = threadIdx.x;
    const float xv = a[(size_t)row*DM_ + tid] + bsrc[(size_t)row*DM_ + tid];

    rs[tid] = xv; __syncthreads();
    for (int s = 64; s > 0; s >>= 1) { if (tid < s) rs[tid] += rs[tid + s]; __syncthreads(); }
    if (tid == 0) stat = rs[0] * (1.0f/DM_);
    __syncthreads();
    const float mean = stat;
    const float d = xv - mean;
    __syncthreads();

    rs[tid] = d * d; __syncthreads();
    for (int s = 64; s > 0; s >>= 1) { if (tid < s) rs[tid] += rs[tid + s]; __syncthreads(); }
    if (tid == 0) stat = rs[0] * (1.0f/DM_);
    __syncthreads();
    const float var = stat;

    out[(size_t)row*DM_ + tid] = d * rsqrtf(var + 1e-5f) * g[tid] + beta[tid];
}

// ---------------------------------------------------------------------------
// 5) Gather last-frame tokens: q [B*256, 128] from det [8192, 128]
// ---------------------------------------------------------------------------
__global__ __launch_bounds__(256)
void copyq_kernel(const float* __restrict__ det, float* __restrict__ q)
{
    const int idx = blockIdx.x * 256 + threadIdx.x;  // 0..131071
    const int b   = idx >> 15;                       // 256*128 per batch
    const int rem = idx & 32767;
    q[idx] = det[(size_t)(b*T_ + (T_-1)) * 256 * 128 + rem];
}

// ---------------------------------------------------------------------------
// 6) Final projection 128->32 with (o*16+bin) scatter: out [64, 512]
// ---------------------------------------------------------------------------
__global__ __launch_bounds__(32)
void proj_kernel(const float* __restrict__ h3, const float* __restrict__ pw,
                 const float* __restrict__ pb, float* __restrict__ out)
{
    const int tok = blockIdx.x;           // 0..1023 = b*256 + n*16 + bin
    const int o   = threadIdx.x;          // 0..31
    const int b   = tok >> 8, loc = tok & 255;
    const int n   = loc >> 4, bin = loc & 15;
    float acc = pb[o];
    const float* hp = h3 + (size_t)tok * DM_;
    const float* wp = pw + (size_t)o   * DM_;
    for (int f = 0; f < DM_; ++f) acc += hp[f] * wp[f];
    out[((size_t)(b*N_ + n))*512 + o*16 + bin] = acc;
}

// ---------------------------------------------------------------------------
extern "C" void kernel_launch(void* const* d_in, const int* in_sizes, int n_in,
                              void* d_out, int out_size, void* d_ws, size_t ws_size,
                              hipStream_t stream)
{
    const float* x       = (const float*)d_in[0];
    const float* bboxes  = (const float*)d_in[1];
    const float* det_w   = (const float*)d_in[2];
    const float* det_b   = (const float*)d_in[3];
    const float* past_w  = (const float*)d_in[4];
    const float* past_b  = (const float*)d_in[5];
    const float* sa_in_w = (const float*)d_in[6];
    const float* sa_in_b = (const float*)d_in[7];
    const float* sa_out_w= (const float*)d_in[8];
    const float* sa_out_b= (const float*)d_in[9];
    const float* ca_in_w = (const float*)d_in[10];
    const float* ca_in_b = (const float*)d_in[11];
    const float* ca_out_w= (const float*)d_in[12];
    const float* ca_out_b= (const float*)d_in[13];
    const float* lin1_w  = (const float*)d_in[14];
    const float* lin1_b  = (const float*)d_in[15];
    const float* lin2_w  = (const float*)d_in[16];
    const float* lin2_b  = (const float*)d_in[17];
    const float* ln1g    = (const float*)d_in[18];
    const float* ln1bb   = (const float*)d_in[19];
    const float* ln2g    = (const float*)d_in[20];
    const float* ln2bb   = (const float*)d_in[21];
    const float* ln3g    = (const float*)d_in[22];
    const float* ln3bb   = (const float*)d_in[23];
    const float* proj_w  = (const float*)d_in[24];
    const float* proj_b  = (const float*)d_in[25];
    (void)in_sizes; (void)n_in; (void)out_size; (void)ws_size;

    float* ws    = (float*)d_ws;
    float* pooled = ws; ws += 8192*256;   // [8192,256]
    float* det    = ws; ws += 8192*128;   // [8192,128]
    float* q      = ws; ws += 1024*128;   // [1024,128]
    float* mem    = ws; ws += 512*128;    // [512,128]
    float* qkv    = ws; ws += 1024*384;   // [1024,384]
    float* attc   = ws; ws += 1024*128;
    float* sao    = ws; ws += 1024*128;
    float* h1     = ws; ws += 1024*128;
    float* qc     = ws; ws += 1024*128;
    float* kvc    = ws; ws += 512*256;    // [512,256] (k|v)
    float* cac    = ws; ws += 1024*128;
    float* cao    = ws; ws += 1024*128;
    float* h2     = ws; ws += 1024*128;
    float* f1     = ws; ws += 1024*512;
    float* f2     = ws; ws += 1024*128;
    float* h3     = ws; ws += 1024*128;

    // 1) ROI pool (HBM-bound pass over x)
    roipool_kernel<<<512, 256, 0, stream>>>(x, bboxes, pooled);

    // 2) det 1x1 conv: [8192,256] x [128,256]^T
    gemm16<0,0><<<(8192/16)*(128/16), 32, 0, stream>>>(pooled, 256, det_w, 256, det_b,
                                                       det, 128, 8192, 128, 256);

    // 3) queries from last frame
    copyq_kernel<<<512, 256, 0, stream>>>(det, q);

    // 4) memory: [512,2048] x past_w(perm)^T + ELU -> [512,128]
    gemm16<2,1><<<(512/16)*(128/16), 32, 0, stream>>>(det, 2048, past_w, 2048, past_b,
                                                      mem, 128, 512, 128, 2048);

    // 5) self-attention
    gemm16<0,0><<<(1024/16)*(384/16), 32, 0, stream>>>(q, 128, sa_in_w, 128, sa_in_b,
                                                       qkv, 384, 1024, 384, 128);
    attn_kernel<<<B_*NHEAD_, 256, 0, stream>>>(qkv, 384, qkv + 128, qkv + 256, 384,
                                               attc, 128, 256);
    gemm16<0,0><<<(1024/16)*(128/16), 32, 0, stream>>>(attc, 128, sa_out_w, 128, sa_out_b,
                                                       sao, 128, 1024, 128, 128);
    add_ln_kernel<<<1024, 128, 0, stream>>>(q, sao, ln1g, ln1bb, h1);

    // 6) cross-attention (K/V from mem)
    gemm16<0,0><<<(1024/16)*(128/16), 32, 0, stream>>>(h1, 128, ca_in_w, 128, ca_in_b,
                                                       qc, 128, 1024, 128, 128);
    gemm16<0,0><<<(512/16)*(256/16), 32, 0, stream>>>(mem, 128, ca_in_w + 128*128, 128,
                                                      ca_in_b + 128, kvc, 256, 512, 256, 128);
    attn_kernel<<<B_*NHEAD_, 256, 0, stream>>>(qc, 128, kvc, kvc + 128, 256,
                                               cac, 128, 128);
    gemm16<0,0><<<(1024/16)*(128/16), 32, 0, stream>>>(cac, 128, ca_out_w, 128, ca_out_b,
                                                       cao, 128, 1024, 128, 128);
    add_ln_kernel<<<1024, 128, 0, stream>>>(h1, cao, ln2g, ln2bb, h2);

    // 7) FFN
    gemm16<1,0><<<(1024/16)*(512/16), 32, 0, stream>>>(h2, 128, lin1_w, 128, lin1_b,
                                                       f1, 512, 1024, 512, 128);
    gemm16<0,0><<<(1024/16)*(128/16), 32, 0, stream>>>(f1, 512, lin2_w, 512, lin2_b,
                                                       f2, 128, 1024, 128, 512);
    add_ln_kernel<<<1024, 128, 0, stream>>>(h2, f2, ln3g, ln3bb, h3);

    // 8) output projection
    proj_kernel<<<1024, 32, 0, stream>>>(h3, proj_w, proj_b, (float*)d_out);
}